// ContrastiveLossPlus_8349416423452
// MI455X (gfx1250) — compile-verified
//
#include <hip/hip_runtime.h>

#define NCLASS 80
#define DDIM   256
#define NROW1  65536
#define NROWS3 (3 * NROW1)
#define EPSF   1e-7f

typedef __attribute__((ext_vector_type(2))) float v2f;
typedef __attribute__((ext_vector_type(8))) float v8f;

#if defined(__gfx1250__) && __has_builtin(__builtin_amdgcn_wmma_f32_16x16x4_f32)
#define HAVE_WMMA_F32X4 1
#else
#define HAVE_WMMA_F32X4 0
#endif

// ---------------------------------------------------------------- zero ws ---
__global__ void supjsd_zero(float* __restrict__ ws, int n) {
  int i = blockIdx.x * 256 + threadIdx.x;
  if (i < n) ws[i] = 0.0f;
}

// ---------------------------------------------------- streaming accumulate ---
// One wave32 per row. Lane owns d = lane + 32*j  (coalesced loads, LDS banks
// conflict-free for the ds_add_f32 sink). Per-WG LDS seg staging (80 KB),
// flushed once with global f32 atomics.
__global__ __launch_bounds__(1024)
void supjsd_accum(const float* __restrict__ a0, const float* __restrict__ a1,
                  const float* __restrict__ a2, const int* __restrict__ labels,
                  float* __restrict__ seg_g, float* __restrict__ segs1_g,
                  int* __restrict__ cnt_g)
{
  __shared__ float seg_lds[NCLASS * DDIM];   // 80 KB
  __shared__ float s1_lds[NCLASS];
  __shared__ int   cnt_lds[NCLASS];

  const int tid = threadIdx.x;
  for (int i = tid; i < NCLASS * DDIM; i += 1024) seg_lds[i] = 0.0f;
  if (tid < NCLASS) { s1_lds[tid] = 0.0f; cnt_lds[tid] = 0; }
  __syncthreads();

  const int lane   = tid & 31;
  const int wave   = tid >> 5;
  const int stride = gridDim.x * 32;  // waves in grid (blockDim = 1024 = 32 waves)

  for (int r = blockIdx.x * 32 + wave; r < NROWS3; r += stride) {
    const int src = r >> 16;             // which of the 3 tensors
    const int row = r & (NROW1 - 1);
    const float* bp = (src == 0) ? a0 : ((src == 1) ? a1 : a2);
    const float* rp = bp + (size_t)row * DDIM;

    // gfx1250 global_prefetch_b8 on next row for this wave
    {
      int r2 = r + stride;
      if (r2 < NROWS3) {
        int s2 = r2 >> 16, w2 = r2 & (NROW1 - 1);
        const float* b2 = (s2 == 0) ? a0 : ((s2 == 1) ? a1 : a2);
        __builtin_prefetch((const void*)((const char*)(b2 + (size_t)w2 * DDIM) + lane * 32), 0, 0);
      }
    }

    float x[8];
#pragma unroll
    for (int j = 0; j < 8; ++j) x[j] = rp[lane + 32 * j];

    float ss = 0.0f;
#pragma unroll
    for (int j = 0; j < 8; ++j) ss += x[j] * x[j];
#pragma unroll
    for (int m = 16; m >= 1; m >>= 1) ss += __shfl_xor(ss, m, 32);

    const float rn  = 1.0f / fmaxf(sqrtf(ss), 1e-12f);
    const int   lab = labels[row];

    float s1 = 0.0f;
#pragma unroll
    for (int j = 0; j < 8; ++j) {
      float pv = x[j] * rn;
      x[j] = pv;
      s1 += (pv > 0.0f) ? pv * __logf(pv) : 0.0f;   // xlogy, 0*log0 = 0
    }
#pragma unroll
    for (int m = 16; m >= 1; m >>= 1) s1 += __shfl_xor(s1, m, 32);

    float* dst = &seg_lds[lab * DDIM];
#pragma unroll
    for (int j = 0; j < 8; ++j) atomicAdd(&dst[lane + 32 * j], x[j]);  // ds_add_f32
    if (lane == 0) { atomicAdd(&s1_lds[lab], s1); atomicAdd(&cnt_lds[lab], 1); }
  }

  __syncthreads();
  for (int i = tid; i < NCLASS * DDIM; i += 1024) atomicAdd(&seg_g[i], seg_lds[i]);
  if (tid < NCLASS) {
    atomicAdd(&segs1_g[tid], s1_lds[tid]);
    atomicAdd(&cnt_g[tid], cnt_lds[tid]);
  }
}

// ------------------------------------------------------------- finalize -----
// loss = 0.01/D * ( sum_c segS1[c]/cnt[c]  -  sum_c mixture_c . log_mix_c )
// The second term: diagonal of the 16x16 class-block Gram Mix @ Lm^T, computed
// with V_WMMA_F32_16X16X4_F32 (fp32 WMMA), 5 class blocks, K = 256.
__global__ __launch_bounds__(256)
void supjsd_finalize(const float* __restrict__ seg_g,
                     const float* __restrict__ segs1_g,
                     const int* __restrict__ cnt_g,
                     float* __restrict__ out)
{
  __shared__ float invden[NCLASS];
  __shared__ float acc;
  __shared__ float tile[5][16][16];

  const int tid = threadIdx.x;
  if (tid == 0) acc = 0.0f;
  if (tid < NCLASS) invden[tid] = 1.0f / fmaxf((float)cnt_g[tid], 1.0f);
  __syncthreads();

  // term1: sum segS1[c]/cnt[c]  (segS1==0 whenever cnt==0, so invden is safe)
  if (tid < NCLASS) atomicAdd(&acc, segs1_g[tid] * invden[tid]);

  const int wave = tid >> 5;
  const int lane = tid & 31;

#if HAVE_WMMA_F32X4
  if (wave < 5) {                         // whole-wave uniform branch: EXEC all 1s
    const int c0 = wave * 16;
    const int m  = lane & 15;             // A: lanes 0-15 / 16-31 both hold M=0..15
    const int kh = (lane >> 4) * 2;       // VGPR0 holds K=kh, VGPR1 holds K=kh+1
    const float inv = invden[c0 + m];
    const float* srow = seg_g + (size_t)(c0 + m) * DDIM;

    v8f cacc = {};
    for (int k0 = 0; k0 < DDIM; k0 += 4) {
      float m0 = srow[k0 + kh]     * inv;       // mixture[c0+m][k]
      float m1 = srow[k0 + kh + 1] * inv;
      v2f a, b;
      a.x = m0;                                 // A = Mix block tile
      a.y = m1;
      b.x = __logf(fmaxf(m0, EPSF));            // B = Lm^T block tile: B[k][n]=lm[c0+n][k]
      b.y = __logf(fmaxf(m1, EPSF));            //   (same lane indexing as A)
      cacc = __builtin_amdgcn_wmma_f32_16x16x4_f32(
          /*neg_a=*/false, a, /*neg_b=*/false, b,
          /*c_mod=*/(short)0, cacc, /*reuse_a=*/false, /*reuse_b=*/false);
    }
    // C/D layout: VGPR j, lanes0-15 -> M=j, N=lane ; lanes16-31 -> M=j+8, N=lane-16
#pragma unroll
    for (int j = 0; j < 8; ++j) {
      int M = j + ((lane >= 16) ? 8 : 0);
      int Nn = lane & 15;
      tile[wave][M][Nn] = cacc[j];
    }
  }
#else
  // scalar fallback: class-diagonal dot only
  if (tid < NCLASS) {
    const float inv = invden[tid];
    const float* srow = seg_g + (size_t)tid * DDIM;
    float d = 0.0f;
    for (int k = 0; k < DDIM; ++k) {
      float mx = srow[k] * inv;
      d += mx * __logf(fmaxf(mx, EPSF));
    }
    tile[tid >> 4][tid & 15][tid & 15] = d;
  }
#endif
  __syncthreads();

  if (tid < NCLASS) {
    int w = tid >> 4, d = tid & 15;
    atomicAdd(&acc, -tile[w][d][d]);    // subtract sum_c mixture_c . lm_c
  }
  __syncthreads();

  if (tid == 0) out[0] = 0.01f * (acc * (1.0f / (float)DDIM));
}

// ------------------------------------------------------------------ launch ---
extern "C" void kernel_launch(void* const* d_in, const int* in_sizes, int n_in,
                              void* d_out, int out_size, void* d_ws, size_t ws_size,
                              hipStream_t stream) {
  const float* a0     = (const float*)d_in[0];
  const float* a1     = (const float*)d_in[1];
  const float* a2     = (const float*)d_in[2];
  const int*   labels = (const int*)d_in[3];

  float* wsf   = (float*)d_ws;
  float* seg   = wsf;                      // 80*256 floats
  float* segs1 = wsf + NCLASS * DDIM;      // 80 floats
  int*   cnt   = (int*)(wsf + NCLASS * DDIM + NCLASS);  // 80 ints
  const int nzero = NCLASS * DDIM + NCLASS + NCLASS;    // 20640 dwords

  supjsd_zero<<<(nzero + 255) / 256, 256, 0, stream>>>(wsf, nzero);
  supjsd_accum<<<256, 1024, 0, stream>>>(a0, a1, a2, labels, seg, segs1, cnt);
  supjsd_finalize<<<1, 256, 0, stream>>>(seg, segs1, cnt, (float*)d_out);
}